// depth_channel_att_60060822667451
// MI455X (gfx1250) — compile-verified
//
#include <hip/hip_runtime.h>
#include <math.h>

// Problem constants (from the reference)
#define BATCH 32
#define CHAN  256
#define HH    64
#define WW    64
#define HW    (HH * WW)           // 4096 floats per (b,c) plane
#define KTAP  3
#define KC    (KTAP * CHAN)       // 768
#define BN_EPS 1e-5f

typedef __attribute__((ext_vector_type(2))) float v2f;
typedef __attribute__((ext_vector_type(8))) float v8f;

// ---------------------------------------------------------------------------
// Kernel 1: global average pool over H*W for each (b,c) plane.
// One 256-thread block per 16KB plane; float4 loads; LDS tree reduction.
// ---------------------------------------------------------------------------
__global__ void gap_kernel(const float* __restrict__ x, float* __restrict__ gap) {
    const int plane = blockIdx.x;                        // b*CHAN + c
    const float4* px = (const float4*)(x + (size_t)plane * HW);
    const int t = threadIdx.x;

    float s = 0.0f;
#pragma unroll
    for (int i = 0; i < 4; ++i) {                        // 1024 float4 / 256 threads
        float4 v = px[i * 256 + t];
        s += (v.x + v.y) + (v.z + v.w);
    }

    __shared__ float red[256];
    red[t] = s;
    __syncthreads();
#pragma unroll
    for (int off = 128; off > 0; off >>= 1) {
        if (t < off) red[t] += red[t + off];
        __syncthreads();
    }
    if (t == 0) gap[plane] = red[0] * (1.0f / (float)HW);
}

// ---------------------------------------------------------------------------
// Kernel 2: f[b,o] = sum_c gap[b,c] * conv_w[o,c]  via V_WMMA_F32_16X16X4_F32,
// then fused BN(eval) + sigmoid epilogue on the accumulator fragment.
// One wave32 per 16x16 output tile: M=32 -> 2 tiles, N=768 -> 48 tiles.
//
// Fragment layouts (CDNA5 ISA 7.12.2):
//   A 16x4 f32: 2 VGPRs; lanes 0-15 hold (K0,K1) of row M=lane,
//               lanes 16-31 hold (K2,K3) of row M=lane-16.
//   B 4x16 f32: 2 VGPRs; N = lane%16; lanes 0-15 hold rows (K0,K1),
//               lanes 16-31 hold rows (K2,K3).
//   C/D 16x16 f32: 8 VGPRs; M = vgpr + 8*(lane>=16), N = lane%16.
// ---------------------------------------------------------------------------
__global__ void filt_gemm_kernel(const float* __restrict__ gap,     // [32, 256]
                                 const float* __restrict__ conv_w,  // [768, 256]
                                 const float* __restrict__ bn_w,
                                 const float* __restrict__ bn_b,
                                 const float* __restrict__ bn_mean,
                                 const float* __restrict__ bn_var,
                                 float* __restrict__ filt)          // [32, 768]
{
    const int lane = threadIdx.x;            // one wave32 per block
    const int tile = blockIdx.x;             // 0..95
    const int m0   = (tile & 1) * 16;        // batch tile
    const int n0   = (tile >> 1) * 16;       // output-channel tile
    const int half = lane >> 4;              // 0 = lanes 0-15, 1 = lanes 16-31
    const int l16  = lane & 15;

    const float* __restrict__ arow = gap    + (size_t)(m0 + l16) * CHAN; // A row M
    const float* __restrict__ bcol = conv_w + (size_t)(n0 + l16) * CHAN; // B col N (conv_w row)

    v8f acc = {};
    for (int k0 = 0; k0 < CHAN; k0 += 4) {
        const int kk = k0 + half * 2;
        v2f a, b;
        a.x = arow[kk];     a.y = arow[kk + 1];
        b.x = bcol[kk];     b.y = bcol[kk + 1];
        // 8 args: (neg_a, A, neg_b, B, c_mod, C, reuse_a, reuse_b)
        acc = __builtin_amdgcn_wmma_f32_16x16x4_f32(
            false, a, false, b, (short)0, acc, false, false);
    }

    // Fused BN (eval mode) + sigmoid, written per the C/D fragment layout.
    const int   n       = n0 + l16;
    const float inv_std = rsqrtf(bn_var[n] + BN_EPS);
    const float w       = bn_w[n];
    const float bb      = bn_b[n];
    const float mu      = bn_mean[n];
#pragma unroll
    for (int i = 0; i < 8; ++i) {
        const int m = m0 + i + half * 8;
        const float g = (acc[i] - mu) * inv_std * w + bb;
        filt[(size_t)m * KC + n] = 1.0f / (1.0f + __expf(-g));
    }
}

// ---------------------------------------------------------------------------
// Kernel 3: fused dynamic 3-tap channel conv (reflect pad) + affine residual.
// out[b,c,hw] = (f0*x[b,cm1] + f1*x[b,c] + f2*x[b,cp1]) * gamma[c] + x[b,c]*beta[c]
// One 256-thread block per (b,c) plane; float4 (B128) loads/stores.
// ---------------------------------------------------------------------------
__global__ void dynconv_kernel(const float* __restrict__ x,
                               const float* __restrict__ filt,   // [32, 768]
                               const float* __restrict__ gamma,  // [256]
                               const float* __restrict__ beta,   // [256]
                               float* __restrict__ out)
{
    const int bc  = blockIdx.x;
    const int b   = bc >> 8;               // / CHAN
    const int c   = bc & (CHAN - 1);
    const int cm1 = (c == 0)        ? 1        : c - 1;   // reflect pad
    const int cp1 = (c == CHAN - 1) ? CHAN - 2 : c + 1;

    const size_t base = (size_t)b * CHAN * HW;
    const float4* __restrict__ x0 = (const float4*)(x + base + (size_t)cm1 * HW);
    const float4* __restrict__ x1 = (const float4*)(x + base + (size_t)c   * HW);
    const float4* __restrict__ x2 = (const float4*)(x + base + (size_t)cp1 * HW);
    float4* __restrict__ o        = (float4*)(out + base + (size_t)c * HW);

    const float f0 = filt[(size_t)b * KC + 0 * CHAN + c];
    const float f1 = filt[(size_t)b * KC + 1 * CHAN + c];
    const float f2 = filt[(size_t)b * KC + 2 * CHAN + c];
    const float gm = gamma[c];
    const float bt = beta[c];

    const int t = threadIdx.x;
#pragma unroll
    for (int i = 0; i < 4; ++i) {                        // 1024 float4 / 256 threads
        const int idx = i * 256 + t;
        const float4 a = x0[idx];
        const float4 m = x1[idx];
        const float4 p = x2[idx];
        float4 r;
        r.x = (f0 * a.x + f1 * m.x + f2 * p.x) * gm + m.x * bt;
        r.y = (f0 * a.y + f1 * m.y + f2 * p.y) * gm + m.y * bt;
        r.z = (f0 * a.z + f1 * m.z + f2 * p.z) * gm + m.z * bt;
        r.w = (f0 * a.w + f1 * m.w + f2 * p.w) * gm + m.w * bt;
        o[idx] = r;
    }
}

// ---------------------------------------------------------------------------
// Launch: three stream-ordered kernels. Workspace: gap (32KB) + filt (96KB).
// ---------------------------------------------------------------------------
extern "C" void kernel_launch(void* const* d_in, const int* in_sizes, int n_in,
                              void* d_out, int out_size, void* d_ws, size_t ws_size,
                              hipStream_t stream) {
    const float* x       = (const float*)d_in[0];   // [32,256,64,64]
    const float* conv_w  = (const float*)d_in[1];   // [768,256]
    const float* bn_w    = (const float*)d_in[2];   // [768]
    const float* bn_b    = (const float*)d_in[3];   // [768]
    const float* bn_mean = (const float*)d_in[4];   // [768]
    const float* bn_var  = (const float*)d_in[5];   // [768]
    const float* gamma   = (const float*)d_in[6];   // [256,1,1]
    const float* beta    = (const float*)d_in[7];   // [256,1,1]
    float* out = (float*)d_out;

    float* gap  = (float*)d_ws;                     // 32*256  floats
    float* filt = gap + BATCH * CHAN;               // 32*768  floats

    // Pass 1: GAP reduction (reads x: 128 MB)
    gap_kernel<<<BATCH * CHAN, 256, 0, stream>>>(x, gap);

    // Pass 2: WMMA GEMM + BN + sigmoid (tiny; 96 waves)
    filt_gemm_kernel<<<(BATCH / 16) * (KC / 16), 32, 0, stream>>>(
        gap, conv_w, bn_w, bn_b, bn_mean, bn_var, filt);

    // Pass 3: fused dynamic conv + affine residual (reads x again — mostly L2
    // resident after pass 1 given 192 MB L2 — and writes out: 128 MB)
    dynconv_kernel<<<BATCH * CHAN, 256, 0, stream>>>(x, filt, gamma, beta, out);
}